// SelfAttention_35648228557123
// MI455X (gfx1250) — compile-verified
//
#include <hip/hip_runtime.h>

typedef __attribute__((ext_vector_type(2))) float v2f;
typedef __attribute__((ext_vector_type(8))) float v8f;

#define NB 4
#define NC 64
#define NN 16384
#define NS 16
#define QSCALE 0.125f   // 64^-0.5

static __device__ __forceinline__ v8f wmma4_f32(v2f a, v2f b, v8f c) {
  // D = A(16x4 f32) * B(4x16 f32) + C(16x16 f32)
  return __builtin_amdgcn_wmma_f32_16x16x4_f32(false, a, false, b, (short)0, c,
                                               false, false);
}

__global__ __launch_bounds__(256) void pt_attn_kernel(
    const float* __restrict__ xyz,        // [B,3,N]
    const float* __restrict__ nxyz,       // [B,3,N,S]
    const float* __restrict__ points,     // [B,C,N]
    const float* __restrict__ npoints,    // [B,C,N,S]
    const float* __restrict__ Wk,         // [C,C]
    const float* __restrict__ Wv,         // [C, C+3]
    const float* __restrict__ Wp,         // [C,4]
    const float* __restrict__ bp,         // [C]
    float* __restrict__ out)              // [B,C,N]
{
  __shared__ float sWk[64 * 64];   // row-major [c_out][c_in]
  __shared__ float sWv[64 * 68];   // row-major, K padded 67->68 (col 67 = 0)
  __shared__ float sWp[64 * 4];
  __shared__ float sBp[64];
  __shared__ float sOut[64 * 8];   // [c][n_in_block]

  const int tid = threadIdx.x;

  // ---- stage weights in LDS (once per block) ----
  for (int i = tid; i < 64 * 64; i += 256) sWk[i] = Wk[i];
  for (int i = tid; i < 64 * 68; i += 256) {
    int o = i / 68, c = i - o * 68;
    sWv[i] = (c < 67) ? Wv[o * 67 + c] : 0.0f;
  }
  sWp[tid & 255] = Wp[tid & 255];
  if (tid < 64) sBp[tid] = bp[tid];
  __syncthreads();

  const int wave = tid >> 5;
  const int lane = tid & 31;
  const int h    = lane >> 4;     // half of the wave (K-offset selector)
  const int s    = lane & 15;     // neighbor column / A-matrix row index
  const int p    = blockIdx.x * 8 + wave;   // global point id
  const int b    = p >> 14;                 // / N
  const int n    = p & (NN - 1);

  // ---- geometry: tmp = xyz - neighbor_xyz, rr = ||tmp|| (per lane's s) ----
  const size_t xyz_base = (size_t)b * 3 * NN + n;
  const float x0 = xyz[xyz_base];
  const float x1 = xyz[xyz_base + NN];
  const float x2 = xyz[xyz_base + 2 * (size_t)NN];
  const size_t nx_base = (((size_t)b * 3) * NN + n) * NS + s;
  const size_t plane   = (size_t)NN * NS;
  const float t0 = x0 - nxyz[nx_base];
  const float t1 = x1 - nxyz[nx_base + plane];
  const float t2 = x2 - nxyz[nx_base + 2 * plane];
  const float rr = sqrtf(t0 * t0 + t1 * t1 + t2 * t2);

  // ---- load neighbor_points into WMMA B-fragment layout; build K-input ----
  // B layout (4x16 fp32): lane l<16 holds B[k0+0..1, l]; lane l+16 holds B[k0+2..3, l]
  float np0[16], np1[16];   // raw neighbor_points (V input rows 0..63)
  float xk0[16], xk1[16];   // neighbor_points + pos_enc (K input)
  const size_t np_base = (((size_t)b * NC) * NN + n) * NS + s;
#pragma unroll
  for (int j = 0; j < 16; ++j) {
    const int c0 = 4 * j + 2 * h;
    const float a0 = npoints[np_base + (size_t)c0 * plane];
    const float a1 = npoints[np_base + (size_t)(c0 + 1) * plane];
    np0[j] = a0;
    np1[j] = a1;
    const float pe0 = sWp[c0 * 4 + 0] * t0 + sWp[c0 * 4 + 1] * t1 +
                      sWp[c0 * 4 + 2] * t2 + sWp[c0 * 4 + 3] * rr + sBp[c0];
    const float pe1 = sWp[(c0 + 1) * 4 + 0] * t0 + sWp[(c0 + 1) * 4 + 1] * t1 +
                      sWp[(c0 + 1) * 4 + 2] * t2 + sWp[(c0 + 1) * 4 + 3] * rr +
                      sBp[c0 + 1];
    xk0[j] = a0 + pe0;
    xk1[j] = a1 + pe1;
  }

  // ---- K projection (Wk @ X) via WMMA, folded directly into logits ----
  // C/D layout: acc[r] holds D[M = 16t + r + 8h, N = s]
  float lg = 0.0f;
  const size_t q_base = ((size_t)b * NC) * NN + n;
#pragma unroll
  for (int t = 0; t < 4; ++t) {
    float qv[8];
#pragma unroll
    for (int r = 0; r < 8; ++r)
      qv[r] = points[q_base + (size_t)(16 * t + 8 * h + r) * NN] * QSCALE;
    v8f acc = {};
#pragma unroll
    for (int j = 0; j < 16; ++j) {
      const int c0 = 4 * j + 2 * h;
      const int arow = 16 * t + s;
      v2f a = {sWk[arow * 64 + c0], sWk[arow * 64 + c0 + 1]};
      v2f bb = {xk0[j], xk1[j]};
      acc = wmma4_f32(a, bb, acc);
    }
#pragma unroll
    for (int r = 0; r < 8; ++r) lg += acc[r] * qv[r];
  }

  // ---- softmax over S=16 (halves hold partial sums over disjoint channels) ----
  lg += __shfl_xor(lg, 16, 32);           // combine channel halves -> full logit[s]
  float mx = lg;
  mx = fmaxf(mx, __shfl_xor(mx, 8, 32));
  mx = fmaxf(mx, __shfl_xor(mx, 4, 32));
  mx = fmaxf(mx, __shfl_xor(mx, 2, 32));
  mx = fmaxf(mx, __shfl_xor(mx, 1, 32));
  float ee = __expf(lg - mx);
  float se = ee;
  se += __shfl_xor(se, 8, 32);
  se += __shfl_xor(se, 4, 32);
  se += __shfl_xor(se, 2, 32);
  se += __shfl_xor(se, 1, 32);
  const float attn = ee / se;             // each lane: attn for its s

  // ---- V projection (Wv @ [np; tmp; 0]) via WMMA, weighted reduce over s ----
#pragma unroll
  for (int t = 0; t < 4; ++t) {
    v8f acc = {};
#pragma unroll
    for (int j = 0; j < 16; ++j) {
      const int c0 = 4 * j + 2 * h;
      const int arow = 16 * t + s;
      v2f a = {sWv[arow * 68 + c0], sWv[arow * 68 + c0 + 1]};
      v2f bb = {np0[j], np1[j]};
      acc = wmma4_f32(a, bb, acc);
    }
    {  // final K-step: rows 64..66 = tmp, row 67 = zero pad
      const int arow = 16 * t + s;
      v2f a = {sWv[arow * 68 + 64 + 2 * h], sWv[arow * 68 + 65 + 2 * h]};
      v2f bb = {h ? t2 : t0, h ? 0.0f : t1};
      acc = wmma4_f32(a, bb, acc);
    }
#pragma unroll
    for (int r = 0; r < 8; ++r) {
      float pv = acc[r] * attn;
      pv += __shfl_xor(pv, 1, 32);
      pv += __shfl_xor(pv, 2, 32);
      pv += __shfl_xor(pv, 4, 32);
      pv += __shfl_xor(pv, 8, 32);        // sum over the 16 neighbors in this half
      if (s == 0) sOut[(16 * t + 8 * h + r) * 8 + wave] = pv;
    }
  }
  __syncthreads();

  // ---- coalesced store: 8 consecutive n per channel row ----
  {
    const int e2 = tid * 2;
    const int c  = e2 >> 3;
    const int nn = e2 & 7;
    const int n0 = (blockIdx.x * 8) & (NN - 1);
    const int bb = (blockIdx.x * 8) >> 14;
    float* dst = out + ((size_t)bb * NC + c) * NN + n0 + nn;
    dst[0] = sOut[c * 8 + nn];
    dst[1] = sOut[c * 8 + nn + 1];
  }
}

extern "C" void kernel_launch(void* const* d_in, const int* in_sizes, int n_in,
                              void* d_out, int out_size, void* d_ws, size_t ws_size,
                              hipStream_t stream) {
  (void)in_sizes; (void)n_in; (void)d_ws; (void)ws_size; (void)out_size;
  const float* xyz      = (const float*)d_in[0];
  const float* nxyz     = (const float*)d_in[1];
  const float* points   = (const float*)d_in[2];
  const float* npoints  = (const float*)d_in[3];
  const float* Wk       = (const float*)d_in[4];
  const float* Wv       = (const float*)d_in[5];
  const float* Wp       = (const float*)d_in[6];
  const float* bp       = (const float*)d_in[7];
  float* out            = (float*)d_out;

  const int blocks = (NB * NN) / 8;   // 8 points (waves) per 256-thread block
  pt_attn_kernel<<<blocks, 256, 0, stream>>>(xyz, nxyz, points, npoints,
                                             Wk, Wv, Wp, bp, out);
}